// SpatialConv_15479062135084
// MI455X (gfx1250) — compile-verified
//
#include <hip/hip_runtime.h>

// CDNA5 / gfx1250 — gathered GEMM via V_WMMA_F32_16X16X4_F32 (full fp32 precision),
// A-tile gathered with GLOBAL_LOAD_ASYNC_TO_LDS_B128 (ASYNCcnt), double-buffered.
// out[b,p,f] = sum_{k,c} (w[p,k] * x[b, idx[p,k], c]) * kernel[k,c,f] + bias[f]
// == GEMM: M = B*N_out = 393216, Kdim = 128 (k*32+c), N = 64.

typedef __attribute__((ext_vector_type(2))) float v2f;
typedef __attribute__((ext_vector_type(8))) float v8f;

// Types matching the async-copy builtin signature: (v4i as1*, v4i as3*, Ii, Ii)
typedef int v4i_b128 __attribute__((vector_size(16)));
typedef __attribute__((address_space(1))) v4i_b128 as1_v4i;
typedef __attribute__((address_space(3))) v4i_b128 as3_v4i;

#define BDIM     8
#define N_IN     196608
#define C_IN     32
#define N_OUT    49152
#define KNB      4
#define FDIM     64
#define KC       128                            // KNB * C_IN, GEMM reduction length
#define TILE_M   16
#define WPB      4                              // waves per block
#define THREADS  (WPB * 32)
#define TILES_TOTAL ((BDIM * N_OUT) / TILE_M)   // 24576
#define NGROUPS  (TILES_TOTAL / WPB)            // 6144
#define GRID     1536
#define GPB      (NGROUPS / GRID)               // 4 groups per block, exact
#define ATILE_F  (TILE_M * KC)                  // 2048 floats = 8KB
#define SMEM_FLOATS (KC * FDIM + WPB * 2 * ATILE_F)   // 96KB total LDS

// ---- CDNA5 async global->LDS copy (16B per lane), ASYNCcnt tracked ----
__device__ __forceinline__ void async_copy_b128(void* lds, const void* g) {
#if __has_builtin(__builtin_amdgcn_global_load_async_to_lds_b128)
    __builtin_amdgcn_global_load_async_to_lds_b128(
        (as1_v4i*)g, (as3_v4i*)lds, 0, 0);
#else
    asm volatile("global_load_async_to_lds_b128 %0, %1, off"
                 :: "v"((unsigned)(unsigned long long)lds),
                    "v"((unsigned long long)g)
                 : "memory");
#endif
}
template <int N>
__device__ __forceinline__ void wait_async_le() {
#if __has_builtin(__builtin_amdgcn_s_wait_asynccnt)
    __builtin_amdgcn_s_wait_asynccnt(N);
#else
    if (N == 0) asm volatile("s_wait_asynccnt 0x0" ::: "memory");
    else        asm volatile("s_wait_asynccnt 0x10" ::: "memory");
#endif
}
__device__ __forceinline__ void wait_ds0() {
    asm volatile("s_wait_dscnt 0x0" ::: "memory");
}

__global__ __launch_bounds__(THREADS)
void spatialconv_wmma_kernel(const float* __restrict__ x,          // (8, 196608, 32)
                             const long long* __restrict__ cidx,   // (49152, 4) int64
                             const float* __restrict__ kw,         // (49152, 4)
                             const float* __restrict__ kern,       // (4, 32, 64) == (128, 64)
                             const float* __restrict__ bias,       // (64,)
                             float* __restrict__ out)              // (8, 49152, 64)
{
    extern __shared__ float smem[];
    float* Bs    = smem;                               // 32 KB, swizzled on kc bit1
    const unsigned tid  = threadIdx.x;
    const unsigned lane = tid & 31u;
    const unsigned wid  = tid >> 5;
    float* Abase = smem + KC * FDIM + wid * (2 * ATILE_F);   // 2 x 8KB per wave

    // Async-gather lane roles: lane covers tap gq (0..3), 16B chunk gch (0..7);
    // one b128 async instruction moves all 4 taps of one output row (512B).
    const unsigned gq  = lane >> 3;
    const unsigned gch = lane & 7u;

    auto issue_gather = [&](float* Atile, unsigned p0, const float* xb) {
        const float* gp[16];
        float*       lp[16];
#pragma unroll
        for (int mm = 0; mm < 16; ++mm) {
            const size_t    pe = (size_t)(p0 + (unsigned)mm) * KNB + gq;
            const long long ni = cidx[pe];
            gp[mm] = xb + (size_t)ni * C_IN + (gch << 2);
            const unsigned cc0 = (gq << 5) + (gch << 2);           // col, mult of 4
            lp[mm] = &Atile[(unsigned)mm * KC + (cc0 ^ ((unsigned)mm << 2))];
        }
#pragma unroll
        for (int mm = 0; mm < 16; ++mm) async_copy_b128(lp[mm], gp[mm]);
    };

    // ---- prologue: start first A gather before staging B (overlap) ----
    unsigned g = blockIdx.x;
    {
        const unsigned row0 = (g * WPB + wid) * TILE_M;
        issue_gather(Abase, row0 % N_OUT,
                     x + (size_t)(row0 / N_OUT) * ((size_t)N_IN * C_IN));
    }

    // ---- stage B = kernel (128 x 64, contiguous) into LDS, float4 chunks ----
    for (unsigned e = tid * 4u; e < KC * FDIM; e += THREADS * 4u) {
        const float4 v = *(const float4*)(kern + e);
        const unsigned kc = e >> 6;
        const unsigned f  = e & 63u;
        const unsigned fs = f ^ (((kc >> 1) & 1u) << 4);           // bank swizzle
        *(float4*)(&Bs[(kc << 6) + fs]) = v;
    }
    __syncthreads();

    // WMMA lane roles (wave32), per ISA VGPR layouts:
    //   A 16x4 f32 : row m = lane&15, K pair offset koff = (lane>>4)*2
    //   B  4x16 f32: same koff, column n = lane&15
    //   C/D 16x16  : VGPR v holds M = v + 8*(lane>>4), N = lane&15 (+16*nt)
    const unsigned m     = lane & 15u;
    const unsigned koff  = (lane >> 4) << 1;
    const unsigned nlane = lane & 15u;
    const unsigned mhi   = (lane >> 4) << 3;
    const unsigned arow  = m * KC;
    const unsigned aswz  = m << 2;

    unsigned buf = 0;
    for (int it = 0; it < GPB; ++it, g += GRID) {
        const unsigned row0 = (g * WPB + wid) * TILE_M;
        const unsigned p0   = row0 % N_OUT;
        float* Acur = Abase + buf * ATILE_F;

        if (it + 1 < GPB) {
            // refill the other buffer while we compute on this one
            const unsigned rn = ((g + GRID) * WPB + wid) * TILE_M;
            wait_ds0();   // WAR: previous compute's LDS reads of that buffer
            issue_gather(Abase + (buf ^ 1u) * ATILE_F, rn % N_OUT,
                         x + (size_t)(rn / N_OUT) * ((size_t)N_IN * C_IN));
            wait_async_le<16>();   // in-order retire => current buffer complete
        } else {
            wait_async_le<0>();
        }

        // per-row tap weights: one aligned float4
        const float4 wt = *(const float4*)(kw + (size_t)(p0 + m) * KNB);

        // accumulators seeded with bias (D = A*B + C chains the K reduction)
        v8f acc[4];
#pragma unroll
        for (int nt = 0; nt < 4; ++nt) {
            const float bv = bias[nt * 16 + nlane];
#pragma unroll
            for (int i = 0; i < 8; ++i) acc[nt][i] = bv;
        }

#pragma unroll
        for (int k = 0; k < 4; ++k) {
            const float wk = (k == 0) ? wt.x : (k == 1) ? wt.y : (k == 2) ? wt.z : wt.w;
#pragma unroll
            for (int j = 0; j < 8; ++j) {
                const unsigned kc = (unsigned)(k * 32 + j * 4) + koff;   // even
                v2f a = *(const v2f*)(&Acur[arow + (kc ^ aswz)]);        // one b64 read
                a.x *= wk;
                a.y *= wk;
                const unsigned bswz  = ((kc >> 1) & 1u) << 4;
                const unsigned bbase = kc << 6;
#pragma unroll
                for (int nt = 0; nt < 4; ++nt) {
                    v2f bv;
                    bv.x = Bs[bbase +      ((nt * 16 + nlane) ^ bswz)];
                    bv.y = Bs[bbase + 64 + ((nt * 16 + nlane) ^ bswz)];
                    acc[nt] = __builtin_amdgcn_wmma_f32_16x16x4_f32(
                        false, a, false, bv, (short)0, acc[nt], false, false);
                }
            }
        }

        // store 16x64 tile; write-once output -> non-temporal stores
        float* orow = out + (size_t)row0 * FDIM + nlane;
#pragma unroll
        for (int vv = 0; vv < 8; ++vv) {
            float* o = orow + (size_t)((unsigned)vv + mhi) * FDIM;
            __builtin_nontemporal_store(acc[0][vv], o);
            __builtin_nontemporal_store(acc[1][vv], o + 16);
            __builtin_nontemporal_store(acc[2][vv], o + 32);
            __builtin_nontemporal_store(acc[3][vv], o + 48);
        }
        buf ^= 1u;
    }
}

extern "C" void kernel_launch(void* const* d_in, const int* in_sizes, int n_in,
                              void* d_out, int out_size, void* d_ws, size_t ws_size,
                              hipStream_t stream) {
    (void)in_sizes; (void)n_in; (void)out_size; (void)d_ws; (void)ws_size;
    const float*     x    = (const float*)d_in[0];
    const long long* cidx = (const long long*)d_in[1];
    const float*     kw   = (const float*)d_in[2];
    const float*     kern = (const float*)d_in[3];
    const float*     bias = (const float*)d_in[4];
    float*           out  = (float*)d_out;

    spatialconv_wmma_kernel<<<GRID, THREADS, SMEM_FLOATS * sizeof(float), stream>>>(
        x, cidx, kw, kern, bias, out);
}